// MPNN5_73469710565617
// MI455X (gfx1250) — compile-verified
//
#include <hip/hip_runtime.h>

typedef __attribute__((ext_vector_type(2))) float v2f;
typedef __attribute__((ext_vector_type(8))) float v8f;

// ---------------------------------------------------------------- utilities
__global__ void zero_kernel(float* __restrict__ p, int n) {
  int i = blockIdx.x * blockDim.x + threadIdx.x;
  if (i < n) p[i] = 0.0f;
}

// deg[col[e]] += ew[e]   (grid-stride, thread per edge)
__global__ void deg_kernel(const long long* __restrict__ col,
                           const float* __restrict__ ew,
                           float* __restrict__ deg, int nE) {
  int i = blockIdx.x * blockDim.x + threadIdx.x;
  int stride = gridDim.x * blockDim.x;
  for (int e = i; e < nE; e += stride)
    atomicAdd(&deg[(int)col[e]], ew[e]);
}

// dinv[i] = rsqrt(deg[i] + 1)   (self-loops add 1; always > 0)
__global__ void dinv_kernel(float* __restrict__ deg, int n) {
  int i = blockIdx.x * blockDim.x + threadIdx.x;
  if (i < n) deg[i] = rsqrtf(deg[i] + 1.0f);
}

// ---------------------------------------------------------------- WMMA GEMM
// H[N,Fout] = X[N,K] @ W[K,Fout], f32, via V_WMMA_F32_16X16X4_F32.
// One wave computes one 16x16 output tile; wave w of the block handles column
// tile w (blockDim.x == 32 * Fout/16). blockIdx.x == row tile (N % 16 == 0).
__global__ void gemm_wmma_f32(const float* __restrict__ X,
                              const float* __restrict__ W,
                              float* __restrict__ H, int K, int Fout) {
  const int lane = threadIdx.x & 31;
  const int wave = threadIdx.x >> 5;
  const int r0   = blockIdx.x * 16;
  const int c0   = wave * 16;
  const int half = lane >> 4;   // 0: lanes 0-15, 1: lanes 16-31
  const int m    = lane & 15;

  v8f acc = {};
  const float* xrow = X + (size_t)(r0 + m) * K;   // A: row m striped in-lane
  for (int k0 = 0; k0 < K; k0 += 4) {
    const int ka = k0 + 2 * half;                 // A VGPR0 holds K=0/2, VGPR1 K=1/3
    v2f a, b;
    a.x = xrow[ka];
    a.y = xrow[ka + 1];
    // B (4x16): VGPR0 = row K=0 (lanes 0-15) / K=2 (lanes 16-31); VGPR1 = K=1/3
    b.x = W[(size_t)ka * Fout + c0 + m];
    b.y = W[(size_t)(ka + 1) * Fout + c0 + m];
    acc = __builtin_amdgcn_wmma_f32_16x16x4_f32(
        /*neg_a=*/false, a, /*neg_b=*/false, b,
        /*c_mod=*/(short)0, acc, /*reuse_a=*/false, /*reuse_b=*/false);
  }
  // D layout: VGPR r -> M=r (lanes 0-15) / M=r+8 (lanes 16-31), N = lane&15
#pragma unroll
  for (int r = 0; r < 8; ++r)
    H[(size_t)(r0 + r + 8 * half) * Fout + c0 + m] = acc[r];
}

// ---------------------------------------------------------------- aggregation
// agg[i, f] = h[i, f] * dinv[i]^2      (self-loop contribution initializes agg)
__global__ void selfloop_kernel(const float* __restrict__ h,
                                const float* __restrict__ dinv,
                                float* __restrict__ agg, int N, int F) {
  int i = blockIdx.x * blockDim.x + threadIdx.x;
  if (i < N * F) {
    int node = i / F;
    float d = dinv[node];
    agg[i] = h[i] * d * d;
  }
}

// Warp-per-edge scatter:  agg[col] += h[row] * (dinv[row]*ew*dinv[col])
// 32 lanes cover the F-wide feature row (coalesced reads + f32 atomics, both
// L2-resident: h and agg each fit easily in the 192 MB L2).
__global__ void edge_scatter_kernel(const long long* __restrict__ row,
                                    const long long* __restrict__ col,
                                    const float* __restrict__ ew,
                                    const float* __restrict__ dinv,
                                    const float* __restrict__ h,
                                    float* __restrict__ agg, int nE, int F) {
  int warp  = (blockIdx.x * blockDim.x + threadIdx.x) >> 5;
  int lane  = threadIdx.x & 31;
  int nwarp = (gridDim.x * blockDim.x) >> 5;
  for (int e = warp; e < nE; e += nwarp) {
    int r = (int)row[e];
    int c = (int)col[e];
    float norm = dinv[r] * ew[e] * dinv[c];
    const float* hr = h + (size_t)r * F;
    float* ac = agg + (size_t)c * F;
    for (int f = lane; f < F; f += 32)
      atomicAdd(&ac[f], hr[f] * norm);
  }
}

// a[i] = act(a[i] + b[i % F])   in place
__global__ void bias_act_kernel(float* __restrict__ a, const float* __restrict__ b,
                                int N, int F, int do_relu) {
  int i = blockIdx.x * blockDim.x + threadIdx.x;
  if (i < N * F) {
    int f = i - (i / F) * F;
    float v = a[i] + b[f];
    a[i] = do_relu ? fmaxf(v, 0.0f) : v;
  }
}

// u[batch[node], f] += agg[node, f] + b2[f]     (global_add_pool, fused bias)
__global__ void pool_kernel(const float* __restrict__ agg,
                            const float* __restrict__ b2,
                            const long long* __restrict__ batch,
                            float* __restrict__ u, int N, int F) {
  int i = blockIdx.x * blockDim.x + threadIdx.x;
  if (i < N * F) {
    int node = i / F;
    int f = i - node * F;
    atomicAdd(&u[(int)batch[node] * F + f], agg[i] + b2[f]);
  }
}

// out[g] = relu(u[g]@Wl1 + bl1) @ Wl2 + bl2     (256 graphs, one thread each)
__global__ void head_kernel(const float* __restrict__ u,
                            const float* __restrict__ Wl1, const float* __restrict__ bl1,
                            const float* __restrict__ Wl2, const float* __restrict__ bl2,
                            float* __restrict__ out, int G) {
  int g = blockIdx.x * blockDim.x + threadIdx.x;
  if (g >= G) return;
  float uin[32];
#pragma unroll
  for (int k = 0; k < 32; ++k) uin[k] = u[g * 32 + k];
  float o = bl2[0];
#pragma unroll
  for (int j = 0; j < 16; ++j) {
    float hs = bl1[j];
#pragma unroll
    for (int k = 0; k < 32; ++k) hs += uin[k] * Wl1[k * 16 + j];
    o += fmaxf(hs, 0.0f) * Wl2[j];
  }
  out[g] = o;
}

// ---------------------------------------------------------------- launcher
extern "C" void kernel_launch(void* const* d_in, const int* in_sizes, int n_in,
                              void* d_out, int out_size, void* d_ws, size_t ws_size,
                              hipStream_t stream) {
  const float*     x     = (const float*)d_in[0];
  const long long* ei    = (const long long*)d_in[1];
  const float*     ew    = (const float*)d_in[2];
  const long long* batch = (const long long*)d_in[3];
  const float* W1  = (const float*)d_in[4];  const float* b1  = (const float*)d_in[5];
  const float* W2  = (const float*)d_in[6];  const float* b2  = (const float*)d_in[7];
  const float* Wl1 = (const float*)d_in[8];  const float* bl1 = (const float*)d_in[9];
  const float* Wl2 = (const float*)d_in[10]; const float* bl2 = (const float*)d_in[11];
  float* out = (float*)d_out;

  const int N = in_sizes[0] / 128;   // 100000
  const int E = in_sizes[1] / 2;     // 3200000
  const int G = out_size;            // 256
  const long long* row = ei;
  const long long* col = ei + E;

  // Workspace layout (floats): dinv | hA (N*64) | hB (N*64) | u (G*32)
  float* ws   = (float*)d_ws;
  float* dinv = ws;
  float* hA   = ws + (((size_t)N + 255) & ~(size_t)255);
  float* hB   = hA + (size_t)N * 64;
  float* u    = hB + (size_t)N * 64;

  const int T = 256;
  // ---- degrees / normalization
  zero_kernel<<<(N + T - 1) / T, T, 0, stream>>>(dinv, N);
  zero_kernel<<<(G * 32 + T - 1) / T, T, 0, stream>>>(u, G * 32);
  deg_kernel<<<2048, T, 0, stream>>>(col, ew, dinv, E);
  dinv_kernel<<<(N + T - 1) / T, T, 0, stream>>>(dinv, N);

  // ---- layer 1: h1 = x @ W1  (K=128, Fout=64), agg1 = out1 in hB
  gemm_wmma_f32<<<N / 16, 32 * (64 / 16), 0, stream>>>(x, W1, hA, 128, 64);
  selfloop_kernel<<<(N * 64 + T - 1) / T, T, 0, stream>>>(hA, dinv, hB, N, 64);
  edge_scatter_kernel<<<4096, T, 0, stream>>>(row, col, ew, dinv, hA, hB, E, 64);
  bias_act_kernel<<<(N * 64 + T - 1) / T, T, 0, stream>>>(hB, b1, N, 64, 1);

  // ---- layer 2: h2 = out1 @ W2  (K=64, Fout=32); h2 in hA[0:N*32], agg2 after
  float* h2   = hA;
  float* agg2 = hA + (size_t)N * 32;
  gemm_wmma_f32<<<N / 16, 32 * (32 / 16), 0, stream>>>(hB, W2, h2, 64, 32);
  selfloop_kernel<<<(N * 32 + T - 1) / T, T, 0, stream>>>(h2, dinv, agg2, N, 32);
  edge_scatter_kernel<<<4096, T, 0, stream>>>(row, col, ew, dinv, h2, agg2, E, 32);

  // ---- pool + head
  pool_kernel<<<(N * 32 + T - 1) / T, T, 0, stream>>>(agg2, b2, batch, u, N, 32);
  head_kernel<<<(G + T - 1) / T, T, 0, stream>>>(u, Wl1, bl1, Wl2, bl2, out, G);
}